// SAGE_processor_29180007809053
// MI455X (gfx1250) — compile-verified
//
#include <hip/hip_runtime.h>

typedef __attribute__((ext_vector_type(2))) float v2f;
typedef __attribute__((ext_vector_type(8))) float v8f;

#define N_NODES_C 100000
#define N_EDGES_C 1600000
#define HIDDEN_C  128
#define EDIM_C    16
#define LDS_STRIDE 132   // 128 + 4 pad: conflict-free strided column reads

// ---------------------------------------------------------------- utilities
__global__ void fill_zero_kernel(float* __restrict__ p, long n) {
    long i = (long)blockIdx.x * blockDim.x + threadIdx.x;
    long stride = (long)gridDim.x * blockDim.x;
    for (; i < n; i += stride) p[i] = 0.0f;
}

__global__ void copy_f32_kernel(const float* __restrict__ in, float* __restrict__ out, long n) {
    long i = (long)blockIdx.x * blockDim.x + threadIdx.x;
    long stride = (long)gridDim.x * blockDim.x;
    for (; i < n; i += stride) out[i] = in[i];
}

__global__ void degree_kernel(const int* __restrict__ dst, float* __restrict__ deg, int nE) {
    int i = blockIdx.x * blockDim.x + threadIdx.x;
    if (i < nE) atomicAdd(&deg[dst[i]], 1.0f);
}

__global__ void invert_deg_kernel(float* __restrict__ deg, int n) {
    int i = blockIdx.x * blockDim.x + threadIdx.x;
    if (i < n) deg[i] = 1.0f / fmaxf(deg[i], 1.0f);
}

// ------------------------------------------------- edge scatter (sum(h[src]) -> agg[dst])
// one lane handles one float4 (1/32 of a 128-float row); 32 lanes per edge.
__global__ void scatter_add_kernel(const float* __restrict__ h,
                                   const int* __restrict__ src,
                                   const int* __restrict__ dst,
                                   float* __restrict__ agg, int nE) {
    long t = (long)blockIdx.x * blockDim.x + threadIdx.x;
    int edge = (int)(t >> 5);
    if (edge >= nE) return;
    int q = (int)(t & 31);
    int s = src[edge];
    int d = dst[edge];
    const float4 v = ((const float4*)(h + (size_t)s * HIDDEN_C))[q];
    float* o = agg + (size_t)d * HIDDEN_C + (size_t)q * 4;
    atomicAdd(o + 0, v.x);
    atomicAdd(o + 1, v.y);
    atomicAdd(o + 2, v.z);
    atomicAdd(o + 3, v.w);
}

// ------------------------------------------------- fused SAGE layer GEMM (WMMA fp32)
// out[n, o] = sum_k h[n,k] Wself[o,k] + (agg[n,:]*inv_deg[n]) . Wneigh[o,:] + b[o]
// Block: 256 threads = 8 waves; block owns 16 node-rows, each wave one 16-col tile.
__global__ __launch_bounds__(256) void sage_gemm_kernel(
    const float* __restrict__ h, const float* __restrict__ agg,
    const float* __restrict__ inv_deg,
    const float* __restrict__ Wself, const float* __restrict__ Wneigh,
    const float* __restrict__ bias, float* __restrict__ out, int do_relu) {

    __shared__ float lh[16 * LDS_STRIDE];
    __shared__ float lg[16 * LDS_STRIDE];

    const int tid  = threadIdx.x;
    const int row0 = blockIdx.x * 16;

    // stage 16x128 tiles of h and scaled agg into LDS (coalesced: rows contiguous)
    for (int idx = tid; idx < 16 * HIDDEN_C; idx += 256) {
        int r = idx >> 7;
        int c = idx & 127;
        long g = (long)(row0 + r) * HIDDEN_C + c;
        lh[r * LDS_STRIDE + c] = h[g];
        lg[r * LDS_STRIDE + c] = agg[g] * inv_deg[row0 + r];
    }
    __syncthreads();

    const int lane = tid & 31;
    const int wave = tid >> 5;          // column tile 0..7
    const int half = lane >> 4;         // 0: K={k,k+1}, 1: K={k+2,k+3}
    const int l16  = lane & 15;

    // B fragment source: B[k][n] = W[(col)*128 + k]  (x @ W^T)
    const float* wsrow = Wself  + (long)(wave * 16 + l16) * HIDDEN_C;
    const float* wnrow = Wneigh + (long)(wave * 16 + l16) * HIDDEN_C;

    v8f acc = {0.f, 0.f, 0.f, 0.f, 0.f, 0.f, 0.f, 0.f};

    for (int k = 0; k < HIDDEN_C; k += 4) {
        const int ka = k + 2 * half;
        v2f a_s, a_n, b_s, b_n;
        // A fragment (16x4): lane l16 = row M, columns ka, ka+1
        a_s.x = lh[l16 * LDS_STRIDE + ka];
        a_s.y = lh[l16 * LDS_STRIDE + ka + 1];
        a_n.x = lg[l16 * LDS_STRIDE + ka];
        a_n.y = lg[l16 * LDS_STRIDE + ka + 1];
        // B fragment (4x16): lane l16 = col N, rows ka, ka+1
        b_s.x = wsrow[ka];
        b_s.y = wsrow[ka + 1];
        b_n.x = wnrow[ka];
        b_n.y = wnrow[ka + 1];
        acc = __builtin_amdgcn_wmma_f32_16x16x4_f32(false, a_s, false, b_s,
                                                    (short)0, acc, false, false);
        acc = __builtin_amdgcn_wmma_f32_16x16x4_f32(false, a_n, false, b_n,
                                                    (short)0, acc, false, false);
    }

    // D layout: VGPR v -> M = v + 8*half ; N = l16
    const int col = wave * 16 + l16;
    const float bb = bias[col];
#pragma unroll
    for (int v = 0; v < 8; ++v) {
        const int m = v + 8 * half;
        float val = acc[v] + bb;
        if (do_relu) val = fmaxf(val, 0.0f);
        out[(long)(row0 + m) * HIDDEN_C + col] = val;
    }
}

// ---------------------------------------------------------------- driver
extern "C" void kernel_launch(void* const* d_in, const int* in_sizes, int n_in,
                              void* d_out, int out_size, void* d_ws, size_t ws_size,
                              hipStream_t stream) {
    const float* h_in   = (const float*)d_in[0];
    const float* e_in   = (const float*)d_in[1];
    const float* Wself  = (const float*)d_in[2];
    const float* Wneigh = (const float*)d_in[3];
    const float* bias   = (const float*)d_in[4];
    const int*   src    = (const int*)d_in[5];
    const int*   dst    = (const int*)d_in[6];
    float* out = (float*)d_out;

    const size_t HN = (size_t)N_NODES_C * HIDDEN_C;

    float* hA  = (float*)d_ws;
    float* hB  = hA + HN;
    float* agg = hB + HN;
    float* deg = agg + HN;

    // degrees -> inverse degrees
    fill_zero_kernel<<<512, 256, 0, stream>>>(deg, (long)N_NODES_C);
    degree_kernel<<<(N_EDGES_C + 255) / 256, 256, 0, stream>>>(dst, deg, N_EDGES_C);
    invert_deg_kernel<<<(N_NODES_C + 255) / 256, 256, 0, stream>>>(deg, N_NODES_C);

    // e passthrough into second output slot
    copy_f32_kernel<<<4096, 256, 0, stream>>>(e_in, out + HN, (long)N_EDGES_C * EDIM_C);

    const float* cur = h_in;
    float* layer_out[3] = { hA, hB, out };
    const long scatter_work = (long)N_EDGES_C * 32;

    for (int layer = 0; layer < 3; ++layer) {
        fill_zero_kernel<<<4096, 256, 0, stream>>>(agg, (long)HN);
        scatter_add_kernel<<<(int)((scatter_work + 255) / 256), 256, 0, stream>>>(
            cur, src, dst, agg, N_EDGES_C);
        sage_gemm_kernel<<<N_NODES_C / 16, 256, 0, stream>>>(
            cur, agg, deg,
            Wself  + (size_t)layer * HIDDEN_C * HIDDEN_C,
            Wneigh + (size_t)layer * HIDDEN_C * HIDDEN_C,
            bias   + (size_t)layer * HIDDEN_C,
            layer_out[layer], layer < 2 ? 1 : 0);
        cur = layer_out[layer];
    }
}